// Knn_32985348833368
// MI455X (gfx1250) — compile-verified
//
#include <hip/hip_runtime.h>
#include <hip/hip_bf16.h>

typedef __attribute__((ext_vector_type(2))) float v2f;
typedef __attribute__((ext_vector_type(8))) float v8f;

#define NPTS  8192
#define KNN   15
#define NTILE (NPTS / 16)
#define BLOCK 64
#define WAVES (BLOCK / 32)

// One wave handles 16 query points against all 8192 candidates.
// Distances come straight out of V_WMMA_F32_16X16X4_F32 via homogeneous
// coordinates: A_j = (xj0, xj1, ||xj||^2, 1), B_i = (-2xi0, -2xi1, 1, ||xi||^2)
// so D[j][i] = ||xi - xj||^2 exactly (f32 FMA accumulation, like the reference).
__global__ __launch_bounds__(BLOCK) void knn_wmma_kernel(
    const float* __restrict__ x, const int* __restrict__ y,
    float* __restrict__ out) {
  __shared__ float mD[WAVES][32][KNN];
  __shared__ float mL[WAVES][32][KNN];

  const int  tid  = threadIdx.x;
  const int  lane = tid & 31;
  const int  wave = tid >> 5;
  const bool hi   = (lane >= 16);
  const int  l16  = lane & 15;

  const int ibase = blockIdx.x * (WAVES * 16) + wave * 16;
  const int i     = ibase + l16;   // this lane's query column

  const v2f* xv = (const v2f*)x;

  // B operand (held constant for the whole loop):
  // lanes 0-15 carry K=0,1 = (-2*xi0, -2*xi1); lanes 16-31 carry K=2,3 = (1, xx_i)
  v2f   qi = xv[i];
  float qq = qi.x * qi.x + qi.y * qi.y;
  v2f b;
  if (!hi) { b.x = -2.0f * qi.x; b.y = -2.0f * qi.y; }
  else     { b.x = 1.0f;         b.y = qq; }

  float dist[KNN];
  float lab[KNN];
#pragma unroll
  for (int s = 0; s < KNN; ++s) { dist[s] = __builtin_inff(); lab[s] = 0.0f; }

  for (int jt = 0; jt < NTILE; ++jt) {
    const int jbase = jt * 16;

    // A operand: lanes 0-15 carry K=0,1 = (xj0, xj1); lanes 16-31 carry
    // K=2,3 = (xx_j, 1) for row j = jbase + (lane & 15).
    const int ja = jbase + l16;
    v2f xj = xv[ja];
    v2f a;
    if (!hi) { a = xj; }
    else     { a.x = xj.x * xj.x + xj.y * xj.y; a.y = 1.0f; }

    v8f c = {0.f, 0.f, 0.f, 0.f, 0.f, 0.f, 0.f, 0.f};
    v8f d = __builtin_amdgcn_wmma_f32_16x16x4_f32(
        /*neg_a=*/false, a, /*neg_b=*/false, b,
        /*c_mod=*/(short)0, c, /*reuse_a=*/false, /*reuse_b=*/false);

    // D layout: lane holds column N = lane&15; VGPR r holds row M = r (+8 for
    // lanes 16-31). So this lane sees 8 candidate distances for query i.
#pragma unroll
    for (int r = 0; r < 8; ++r) {
      const int j  = jbase + r + (hi ? 8 : 0);
      float     dd = d[r];
      // exclude self; cheap guarded sorted-insert (rare after warm-up)
      if (j != i && dd < dist[KNN - 1]) {
        float lv = (float)y[j];
#pragma unroll
        for (int s = 0; s < KNN; ++s) {
          bool  take = dd < dist[s];
          float td = dist[s], tl = lab[s];
          dist[s] = take ? dd : td;
          lab[s]  = take ? lv : tl;
          dd = take ? td : dd;
          lv = take ? tl : lv;
        }
      }
    }
  }

  // Lanes l and l+16 each hold a sorted top-15 for query ibase + (l&15).
  // Merge the two sorted lists (exact top-15 of the union) and vote.
#pragma unroll
  for (int s = 0; s < KNN; ++s) {
    mD[wave][lane][s] = dist[s];
    mL[wave][lane][s] = lab[s];
  }
  __syncthreads();

  if (!hi) {
    const float* dA = mD[wave][lane];
    const float* lA = mL[wave][lane];
    const float* dB = mD[wave][lane + 16];
    const float* lB = mL[wave][lane + 16];
    int pa = 0, pb = 0;
    float sum = 0.0f;
#pragma unroll
    for (int s = 0; s < KNN; ++s) {
      bool ta = dA[pa] <= dB[pb];
      sum += ta ? lA[pa] : lB[pb];
      pa += ta ? 1 : 0;
      pb += ta ? 0 : 1;
    }
    out[i] = (sum > 7.5f) ? 1.0f : 0.0f;   // majority of 15 binary labels
  }
}

extern "C" void kernel_launch(void* const* d_in, const int* in_sizes, int n_in,
                              void* d_out, int out_size, void* d_ws, size_t ws_size,
                              hipStream_t stream) {
  (void)in_sizes; (void)n_in; (void)out_size; (void)d_ws; (void)ws_size;
  const float* x  = (const float*)d_in[0];   // [8192,2] f32
  const int*   y  = (const int*)d_in[1];     // [8192]   i32
  float*       out = (float*)d_out;          // [8192]   f32

  dim3 grid(NPTS / (WAVES * 16));  // 256 blocks
  dim3 block(BLOCK);               // 64 threads = 2 waves
  knn_wmma_kernel<<<grid, block, 0, stream>>>(x, y, out);
}